// GPool_90486370992981
// MI455X (gfx1250) — compile-verified
//
#include <hip/hip_runtime.h>
#include <hip/hip_bf16.h>
#include <math.h>

#define N_NODES 10000
#define F_DIM   128
#define E_EDGES 320000
#define K_POOL  5000   // RATIO * N

typedef float v2f __attribute__((ext_vector_type(2)));
typedef float v8f __attribute__((ext_vector_type(8)));

// ---------------------------------------------------------------------------
// Kernel A: invnorm = 1/||w||  (single block of 128 threads)
// ---------------------------------------------------------------------------
__global__ void wnorm_kernel(const float* __restrict__ w, float* __restrict__ invnorm) {
    __shared__ float red[128];
    int t = threadIdx.x;
    float v = w[t];
    red[t] = v * v;
    __syncthreads();
    for (int s = 64; s > 0; s >>= 1) {
        if (t < s) red[t] += red[t + s];
        __syncthreads();
    }
    if (t == 0) *invnorm = rsqrtf(red[0]);
}

// ---------------------------------------------------------------------------
// Kernel B: normalized scores via V_WMMA_F32_16X16X4_F32.
// One wave handles a 16-row tile of xl. B replicates w across all 16 columns,
// so every column of D holds the same 16 scores; lanes 0 and 16 write them out.
// A layout (32-bit 16x4): lanes 0-15 = M, VGPR0={K0|K2 by half}, VGPR1={K1|K3}.
// C/D layout: VGPR v, lanes 0-15 -> M=v ; lanes 16-31 -> M=v+8.
// ---------------------------------------------------------------------------
__global__ void score_kernel(const float* __restrict__ xl, const float* __restrict__ w,
                             const float* __restrict__ invnorm_p, float* __restrict__ y) {
    const int lane = threadIdx.x & 31;
    const int wave = (blockIdx.x * blockDim.x + threadIdx.x) >> 5;
    const int numTiles = N_NODES / 16;          // 625 exactly
    if (wave >= numTiles) return;               // wave-uniform guard: EXEC all-ones below
    const int base = wave * 16;
    const int half = lane >> 4;                 // 0: K{0,1}, 1: K{2,3}
    const int lr   = lane & 15;                 // row within tile (for A)
    const float inv = *invnorm_p;

#if __has_builtin(__builtin_amdgcn_wmma_f32_16x16x4_f32)
    v8f c = {};
    const float* arow = xl + (size_t)(base + lr) * F_DIM + 2 * half;
    const float* wp   = w + 2 * half;
    #pragma unroll 4
    for (int kk = 0; kk < F_DIM / 4; ++kk) {
        v2f a = { arow[kk * 4], arow[kk * 4 + 1] };
        v2f b = { wp[kk * 4],   wp[kk * 4 + 1]   };   // replicated over all N columns
        c = __builtin_amdgcn_wmma_f32_16x16x4_f32(
                /*neg_a=*/false, a, /*neg_b=*/false, b,
                /*c_mod=*/(short)0, c, /*reuse_a=*/false, /*reuse_b=*/false);
    }
    if (lr == 0) {      // lane 0 -> M=0..7, lane 16 -> M=8..15
        #pragma unroll
        for (int v = 0; v < 8; ++v)
            y[base + 8 * half + v] = c[v] * inv;
    }
#else
    if (lane < 16) {
        const float* arow = xl + (size_t)(base + lane) * F_DIM;
        float acc = 0.f;
        for (int f = 0; f < F_DIM; ++f) acc += arow[f] * w[f];
        y[base + lane] = acc * inv;
    }
#endif
}

// ---------------------------------------------------------------------------
// Kernel C: exact stable top-k by rank counting. y fits in LDS (40 KB).
// rank(i) = #{ j : y_j > y_i  or (y_j == y_i and j < i) }  -> permutation,
// identical ordering semantics to lax.top_k (descending, ties by lower index).
// ---------------------------------------------------------------------------
__global__ void rank_kernel(const float* __restrict__ y, int* __restrict__ pos,
                            int* __restrict__ idx, float* __restrict__ out_idx_f) {
    __shared__ float sy[N_NODES];               // 40 KB of the 320 KB WGP LDS
    for (int j = threadIdx.x; j < N_NODES; j += blockDim.x) sy[j] = y[j];
    __syncthreads();
    int i = blockIdx.x * blockDim.x + threadIdx.x;
    if (i >= N_NODES) return;
    const float yi = sy[i];
    int rank = 0;
    for (int j = 0; j < N_NODES; ++j) {
        float yj = sy[j];
        rank += (yj > yi) || (yj == yi && j < i);
    }
    if (rank < K_POOL) {
        idx[rank]       = i;
        pos[i]          = rank;
        out_idx_f[rank] = (float)i;             // top_idxs output (float dtype in d_out)
    } else {
        pos[i] = -1;
    }
}

// ---------------------------------------------------------------------------
// Kernel D: xl1[r,:] = xl[idx[r],:] * sigmoid(y[idx[r]])   (float4 vectorized)
// ---------------------------------------------------------------------------
__global__ void gather_kernel(const float* __restrict__ xl, const float* __restrict__ y,
                              const int* __restrict__ idx, float* __restrict__ xl1) {
    const int F4 = F_DIM / 4;
    int t = blockIdx.x * blockDim.x + threadIdx.x;
    if (t >= K_POOL * F4) return;
    int r  = t / F4;
    int f4 = t % F4;
    int node = idx[r];
    float gate = 1.0f / (1.0f + __expf(-y[node]));
    float4 v = ((const float4*)(xl + (size_t)node * F_DIM))[f4];
    v.x *= gate; v.y *= gate; v.z *= gate; v.w *= gate;
    ((float4*)(xl1 + (size_t)r * F_DIM))[f4] = v;
}

// ---------------------------------------------------------------------------
// Kernel E: zero the 100 MB pooled adjacency (the bandwidth floor of this op).
// ---------------------------------------------------------------------------
__global__ void fill0_kernel(float4* __restrict__ p, long n4) {
    long i = (long)blockIdx.x * blockDim.x + threadIdx.x;
    long stride = (long)gridDim.x * blockDim.x;
    float4 z = {0.f, 0.f, 0.f, 0.f};
    for (; i < n4; i += stride) p[i] = z;
}

// ---------------------------------------------------------------------------
// Kernel F: scatter edges directly into al1 (skips reference's 400 MB dense AL).
// Sum of 1.0f contributions is exact -> deterministic despite atomics.
// ---------------------------------------------------------------------------
__global__ void edge_kernel(const int* __restrict__ ei, const int* __restrict__ pos,
                            float* __restrict__ al1) {
    int e = blockIdx.x * blockDim.x + threadIdx.x;
    if (e >= E_EDGES) return;
    int u = ei[e];
    int v = ei[E_EDGES + e];
    int ru = pos[u];
    int rv = pos[v];
    if (ru >= 0 && rv >= 0)
        atomicAdd(al1 + (long)ru * K_POOL + rv, 1.0f);
}

// ---------------------------------------------------------------------------
extern "C" void kernel_launch(void* const* d_in, const int* in_sizes, int n_in,
                              void* d_out, int out_size, void* d_ws, size_t ws_size,
                              hipStream_t stream) {
    const float* xl = (const float*)d_in[0];          // [10000,128]
    const int*   ei = (const int*)d_in[1];            // [2,320000]
    const float* w  = (const float*)d_in[2];          // [1,128]

    float* out      = (float*)d_out;
    float* out_xl1  = out;                            // [5000,128]
    float* out_al1  = out + (size_t)K_POOL * F_DIM;   // [5000,5000]
    float* out_tidx = out_al1 + (size_t)K_POOL * K_POOL; // [5000]

    float* ws_y    = (float*)d_ws;                    // [10000] normalized scores
    int*   ws_pos  = (int*)(ws_y + N_NODES);          // [10000] node -> rank or -1
    int*   ws_idx  = ws_pos + N_NODES;                // [5000]  rank -> node
    float* ws_inv  = (float*)(ws_idx + K_POOL);       // [1]     1/||w||

    // A: weight norm
    wnorm_kernel<<<1, 128, 0, stream>>>(w, ws_inv);

    // B: WMMA scores (625 waves = 16-row tiles; 8 waves / 256-thread block)
    {
        int waves = N_NODES / 16;
        int blocks = (waves + 7) / 8;
        score_kernel<<<blocks, 256, 0, stream>>>(xl, w, ws_inv, ws_y);
    }

    // C: stable top-k via rank counting
    rank_kernel<<<(N_NODES + 255) / 256, 256, 0, stream>>>(ws_y, ws_pos, ws_idx, out_tidx);

    // D: gated gather of selected rows
    gather_kernel<<<(K_POOL * (F_DIM / 4) + 255) / 256, 256, 0, stream>>>(xl, ws_y, ws_idx, out_xl1);

    // E: zero pooled adjacency (100 MB -> the op's bandwidth floor)
    {
        long n4 = (long)K_POOL * K_POOL / 4;
        fill0_kernel<<<4096, 256, 0, stream>>>((float4*)out_al1, n4);
    }

    // F: scatter surviving edges
    edge_kernel<<<(E_EDGES + 255) / 256, 256, 0, stream>>>(ei, ws_pos, out_al1);
}